// SAGE_20237885899316
// MI455X (gfx1250) — compile-verified
//
#include <hip/hip_runtime.h>

typedef __attribute__((ext_vector_type(2))) float v2f;
typedef __attribute__((ext_vector_type(8))) float v8f;

#define FEATS 128
#define TILE_M 16

// ---------------- zero workspace ----------------
__global__ void sage_zero_kernel(float* __restrict__ p, long long n) {
  long long i = (long long)blockIdx.x * blockDim.x + threadIdx.x;
  long long stride = (long long)gridDim.x * blockDim.x;
  for (; i < n; i += stride) p[i] = 0.0f;
}

// ---------------- edge scatter: msg[dst] += feat[src]; deg[dst] += 1 ----------------
// One wave32 per edge; each lane moves 4 floats (4*32 = 128 features).
__global__ void sage_scatter_kernel(const float* __restrict__ feat,
                                    const int* __restrict__ src,
                                    const int* __restrict__ dst,
                                    float* __restrict__ msg,
                                    float* __restrict__ deg,
                                    int n_edges, int count_deg) {
  int gid  = blockIdx.x * blockDim.x + threadIdx.x;
  int edge = gid >> 5;
  int lane = threadIdx.x & 31;
  if (edge >= n_edges) return;
  int s = src[edge];
  int d = dst[edge];
  const float4 v = *(const float4*)(feat + (long long)s * FEATS + lane * 4);
  float* mp = msg + (long long)d * FEATS + lane * 4;
  atomicAdd(mp + 0, v.x);
  atomicAdd(mp + 1, v.y);
  atomicAdd(mp + 2, v.z);
  atomicAdd(mp + 3, v.w);
  if (count_deg && lane == 0) atomicAdd(deg + d, 1.0f);
}

// ---------------- fused normalize + GEMM(WMMA f32) + bias (+ReLU) ----------------
// Block: (NCOLS/16) waves. Each block handles a 16-node tile; each wave one
// 16x16 output tile. hbar staged in LDS (row pitch 132 to dodge bank conflicts).
template <int NCOLS, bool RELU>
__global__ void sage_layer_kernel(const float* __restrict__ feat,  // [N,128]
                                  const float* __restrict__ msg,   // [N,128]
                                  const float* __restrict__ deg,   // [N]
                                  const float* __restrict__ W,     // [128,NCOLS] row-major
                                  const float* __restrict__ bias,  // [NCOLS]
                                  float* __restrict__ out,         // [N,NCOLS]
                                  int n_nodes) {
  __shared__ float hbar[TILE_M][FEATS + 4];

  const int base = blockIdx.x * TILE_M;
  const int tid  = threadIdx.x;
  const int nthreads = 32 * (NCOLS / 16);

  // Cooperative: hbar[m][k] = (msg + feat) / (deg + 1)
  for (int idx = tid; idx < TILE_M * FEATS; idx += nthreads) {
    int m = idx >> 7;             // / 128
    int k = idx & (FEATS - 1);
    int node = base + m;
    if (node >= n_nodes) node = n_nodes - 1;   // clamp (tail-safe; 100000%16==0 anyway)
    float r = 1.0f / (deg[node] + 1.0f);
    hbar[m][k] = (msg[(long long)node * FEATS + k] + feat[(long long)node * FEATS + k]) * r;
  }
  __syncthreads();

  const int wave = tid >> 5;
  const int lane = tid & 31;
  const int col  = lane & 15;     // A: M index ; B/D: N index within tile
  const int half = lane >> 4;     // 0: K+0/K+1 ; 1: K+2/K+3  (ISA 16x4 f32 A layout)
  const int n0   = wave * 16;

  v8f acc = {};
#pragma unroll 8
  for (int k = 0; k < FEATS; k += 4) {
    v2f a, b;
    a.x = hbar[col][k + 2 * half];
    a.y = hbar[col][k + 2 * half + 1];
    b.x = W[(k + 2 * half) * NCOLS + n0 + col];
    b.y = W[(k + 2 * half + 1) * NCOLS + n0 + col];
    // D = A(16x4,f32) * B(4x16,f32) + C  — v_wmma_f32_16x16x4_f32
    acc = __builtin_amdgcn_wmma_f32_16x16x4_f32(false, a, false, b,
                                                (short)0, acc, false, false);
  }

  const float bv = bias[n0 + col];
  // D layout: VGPR r holds row r (lanes 0-15) and row r+8 (lanes 16-31).
  // Per-lane base address for row (base + 8*half); row r is an immediate offset.
  float* obase = out + ((long long)(base + 8 * half) * NCOLS + n0 + col);

  if (base + TILE_M <= n_nodes) {
    // Full tile: straight-line stores, no exec-mask churn.
#pragma unroll
    for (int r = 0; r < 8; ++r) {
      float v = acc[r] + bv;
      if (RELU) v = fmaxf(v, 0.0f);
      obase[(long long)r * NCOLS] = v;
    }
  } else {
    // Tail tile (unused when n_nodes % 16 == 0): guarded stores.
#pragma unroll
    for (int r = 0; r < 8; ++r) {
      int node = base + r + 8 * half;
      if (node < n_nodes) {
        float v = acc[r] + bv;
        if (RELU) v = fmaxf(v, 0.0f);
        obase[(long long)r * NCOLS] = v;
      }
    }
  }
}

extern "C" void kernel_launch(void* const* d_in, const int* in_sizes, int n_in,
                              void* d_out, int out_size, void* d_ws, size_t ws_size,
                              hipStream_t stream) {
  const float* x  = (const float*)d_in[0];
  const int*   src = (const int*)d_in[1];
  const int*   dst = (const int*)d_in[2];
  const float* W1 = (const float*)d_in[3];
  const float* b1 = (const float*)d_in[4];
  const float* W2 = (const float*)d_in[5];
  const float* b2 = (const float*)d_in[6];
  float* out = (float*)d_out;

  const int n_edges = in_sizes[1];
  const int n_nodes = in_sizes[0] / FEATS;

  // Workspace layout: msg [N*128] | deg [N] | h1 [N*128]  (~103 MB)
  float* msg = (float*)d_ws;
  float* deg = msg + (long long)n_nodes * FEATS;
  float* h1  = deg + n_nodes;

  const long long msg_elems = (long long)n_nodes * FEATS;
  const int scat_blocks = (n_edges * 32 + 255) / 256;
  const int tile_blocks = (n_nodes + TILE_M - 1) / TILE_M;

  // ---- Layer 1 ----
  sage_zero_kernel<<<2048, 256, 0, stream>>>(msg, msg_elems + n_nodes);  // msg + deg
  sage_scatter_kernel<<<scat_blocks, 256, 0, stream>>>(x, src, dst, msg, deg, n_edges, 1);
  sage_layer_kernel<128, true><<<tile_blocks, 256, 0, stream>>>(x, msg, deg, W1, b1, h1, n_nodes);

  // ---- Layer 2 (reuse msg; deg unchanged) ----
  sage_zero_kernel<<<2048, 256, 0, stream>>>(msg, msg_elems);
  sage_scatter_kernel<<<scat_blocks, 256, 0, stream>>>(h1, src, dst, msg, deg, n_edges, 0);
  sage_layer_kernel<64, false><<<tile_blocks, 128, 0, stream>>>(h1, msg, deg, W2, b2, out, n_nodes);
}